// ProposalLayer_1675037246050
// MI455X (gfx1250) — compile-verified
//
#include <hip/hip_runtime.h>
#include <stdint.h>

#define KPRE   6000     // PRE_NMS top-k
#define PROP   1000     // post-NMS proposals
#define SORTN  8192     // padded sort size (pow2 >= KPRE + slack)
#define NBKT   2048     // 11-bit radix buckets
#define NMS_T  0.7f
#define MWORDS ((KPRE + 31) / 32)
#define BOXF   (KPRE * 4)          // floats per image box tile (24000)

typedef unsigned uv4 __attribute__((ext_vector_type(4)));
typedef unsigned uv8 __attribute__((ext_vector_type(8)));

// Monotonic float->uint key (scores are softmax probs > 0, but keep it general).
__device__ __forceinline__ unsigned f2key(float f) {
    unsigned u = __float_as_uint(f);
    return (u & 0x80000000u) ? ~u : (u | 0x80000000u);
}

// ---------------------------------------------------------------- zero scratch
__global__ void k_zero(unsigned* p, size_t nwords) {
    size_t i = (size_t)blockIdx.x * blockDim.x + threadIdx.x;
    if (i < nwords) p[i] = 0u;
}

// -------------------------------------------------- level-1 histogram (11 bit)
__global__ void k_hist1(const float* __restrict__ probs, unsigned* __restrict__ hist1, int N) {
    __shared__ unsigned lh[NBKT];
    int b = blockIdx.y;
    for (int i = threadIdx.x; i < NBKT; i += blockDim.x) lh[i] = 0u;
    __syncthreads();
    int n = blockIdx.x * blockDim.x + threadIdx.x;
    if (n < N) {
        size_t off = ((size_t)b * N + n) * 2 + 1;
        // stream-ahead prefetch (emits global_prefetch_b8)
        __builtin_prefetch(probs + off + 2u * blockDim.x, 0, 0);
        unsigned key = f2key(probs[off]);
        atomicAdd(&lh[key >> 21], 1u);
    }
    __syncthreads();
    for (int i = threadIdx.x; i < NBKT; i += blockDim.x)
        if (lh[i]) atomicAdd(&hist1[(size_t)b * NBKT + i], lh[i]);
}

// ctr layout per image (8 u32): [0]=t1 [1]=above1 [2]=t2 [3]=above2 [4]=cntA [5]=cntE
__global__ void k_select1(const unsigned* __restrict__ hist1, unsigned* __restrict__ ctrs) {
    int b = blockIdx.x;
    if (threadIdx.x != 0) return;
    unsigned cum = 0, above = 0; int t = 0;
    for (int bk = NBKT - 1; bk >= 0; --bk) {
        unsigned c = hist1[(size_t)b * NBKT + bk];
        if (cum + c >= KPRE || bk == 0) { t = bk; above = cum; break; }
        cum += c;
    }
    ctrs[b * 8 + 0] = (unsigned)t;
    ctrs[b * 8 + 1] = above;
}

// ------------------------------------------- level-2 histogram (next 11 bits)
__global__ void k_hist2(const float* __restrict__ probs, const unsigned* __restrict__ ctrs,
                        unsigned* __restrict__ hist2, int N) {
    int b = blockIdx.y;
    unsigned t1 = ctrs[b * 8 + 0];
    int n = blockIdx.x * blockDim.x + threadIdx.x;
    if (n >= N) return;
    unsigned key = f2key(probs[((size_t)b * N + n) * 2 + 1]);
    if ((key >> 21) == t1)
        atomicAdd(&hist2[(size_t)b * NBKT + ((key >> 10) & 2047u)], 1u);
}

__global__ void k_select2(const unsigned* __restrict__ hist2, unsigned* __restrict__ ctrs) {
    int b = blockIdx.x;
    if (threadIdx.x != 0) return;
    unsigned above1 = ctrs[b * 8 + 1];
    unsigned need1  = (above1 < KPRE) ? (KPRE - above1) : 0u;
    unsigned cum = 0, above2 = 0; int t2 = 0;
    for (int bk = NBKT - 1; bk >= 0; --bk) {
        unsigned c = hist2[(size_t)b * NBKT + bk];
        if (cum + c >= need1 || bk == 0) { t2 = bk; above2 = cum; break; }
        cum += c;
    }
    ctrs[b * 8 + 2] = (unsigned)t2;
    ctrs[b * 8 + 3] = above2;
    ctrs[b * 8 + 4] = 0u;   // cntA
    ctrs[b * 8 + 5] = 0u;   // cntE
}

// ------------------------------------------------------------- compact top-k
// pairs entry: (key << 32) | (~idx)  -> descending sort gives smallest idx first on ties
__global__ void k_compact(const float* __restrict__ probs, unsigned* __restrict__ ctrs,
                          unsigned long long* __restrict__ pairs, int N) {
    int b = blockIdx.y;
    unsigned t1 = ctrs[b * 8 + 0], above1 = ctrs[b * 8 + 1];
    unsigned t2 = ctrs[b * 8 + 2], above2 = ctrs[b * 8 + 3];
    int n = blockIdx.x * blockDim.x + threadIdx.x;
    if (n >= N) return;
    unsigned key = f2key(probs[((size_t)b * N + n) * 2 + 1]);
    unsigned b1 = key >> 21, b2 = (key >> 10) & 2047u;
    unsigned long long pk = ((unsigned long long)key << 32) | (0xFFFFFFFFu - (unsigned)n);
    if (b1 > t1 || (b1 == t1 && b2 > t2)) {
        unsigned p = atomicAdd(&ctrs[b * 8 + 4], 1u);
        if (p < SORTN) pairs[(size_t)b * SORTN + p] = pk;
    } else if (b1 == t1 && b2 == t2) {
        unsigned e = atomicAdd(&ctrs[b * 8 + 5], 1u);
        unsigned p = above1 + above2 + e;   // take whole tie bucket (up to capacity); sort fixes order
        if (p < SORTN) pairs[(size_t)b * SORTN + p] = pk;
    }
}

// ------------------------------- bitonic sort (desc) + box decode/clip to ws
// Pairs tile (64 KB) staged into LDS via ASYNCcnt-tracked async DMA.
__global__ void k_sort_decode(const unsigned long long* __restrict__ pairs,
                              const float* __restrict__ deltas,
                              const float* __restrict__ anchors,
                              float* __restrict__ boxes, int N) {
    extern __shared__ unsigned long long sp[];  // SORTN u64 = 64 KB dynamic LDS
    int b = blockIdx.x, tid = threadIdx.x, bd = blockDim.x;

    const unsigned long long* gsrc = pairs + (size_t)b * SORTN;
    unsigned ldsBase = (unsigned)(uintptr_t)sp;
    for (int t = tid; t < SORTN / 2; t += bd) {        // 16 B (2 pairs) per lane-op
        unsigned off = (unsigned)t * 16u;
        asm volatile("global_load_async_to_lds_b128 %0, %1, %2"
                     :: "v"(ldsBase + off), "v"(off), "s"(gsrc)
                     : "memory");
    }
    asm volatile("s_wait_asynccnt 0" ::: "memory");
    __syncthreads();

    for (unsigned k = 2; k <= SORTN; k <<= 1) {
        for (unsigned j = k >> 1; j > 0; j >>= 1) {
            for (unsigned i = tid; i < SORTN; i += (unsigned)bd) {
                unsigned ij = i ^ j;
                if (ij > i) {
                    unsigned long long a = sp[i], c = sp[ij];
                    bool desc = ((i & k) == 0);
                    if (desc ? (a < c) : (a > c)) { sp[i] = c; sp[ij] = a; }
                }
            }
            __syncthreads();
        }
    }
    for (int jj = tid; jj < KPRE; jj += bd) {
        unsigned long long v = sp[jj];
        unsigned key = (unsigned)(v >> 32);
        float o0 = 0.f, o1 = 0.f, o2 = 0.f, o3 = 0.f;
        if (key != 0u) {
            unsigned idx = 0xFFFFFFFFu - (unsigned)(v & 0xFFFFFFFFull);
            size_t base = ((size_t)b * N + idx) * 4;
            float ay1 = anchors[base + 0], ax1 = anchors[base + 1];
            float ay2 = anchors[base + 2], ax2 = anchors[base + 3];
            float dy = deltas[base + 0] * 0.1f, dx = deltas[base + 1] * 0.1f;
            float dh = deltas[base + 2] * 0.2f, dw = deltas[base + 3] * 0.2f;
            float h = ay2 - ay1, w = ax2 - ax1;
            float cy = ay1 + 0.5f * h + dy * h;
            float cx = ax1 + 0.5f * w + dx * w;
            h *= __expf(dh); w *= __expf(dw);
            float y1 = cy - 0.5f * h, x1 = cx - 0.5f * w;
            float y2 = y1 + h,        x2 = x1 + w;
            o0 = fminf(fmaxf(y1, 0.f), 1.f);
            o1 = fminf(fmaxf(x1, 0.f), 1.f);
            o2 = fminf(fmaxf(y2, 0.f), 1.f);
            o3 = fminf(fmaxf(x2, 0.f), 1.f);
        }
        size_t o = ((size_t)b * KPRE + jj) * 4;
        boxes[o + 0] = o0; boxes[o + 1] = o1; boxes[o + 2] = o2; boxes[o + 3] = o3;
    }
}

// ------------------------------------------------------ greedy NMS per image
// Box tile (96 KB) staged into LDS by the Tensor Data Mover: one wave issues a
// single tensor_load_to_lds with a 2-group D# (1D tile, 24000 x 4B elements),
// then waits on TENSORcnt.
__global__ void k_nms(const float* __restrict__ boxesAll, float* __restrict__ out) {
    extern __shared__ float sm[];           // [KPRE*4] boxes + [KPRE] areas = 120000 B
    float* sb = sm;
    float* sa = sm + KPRE * 4;
    __shared__ unsigned mask[MWORDS];
    __shared__ unsigned short keptIdx[PROP];

    int b = blockIdx.x, tid = threadIdx.x, bd = blockDim.x;
    const float* gsrc = boxesAll + (size_t)b * BOXF;

    if (tid < 32) {                                    // wave 0 drives the TDM
        unsigned long long ga = (unsigned long long)(uintptr_t)gsrc;
        unsigned ldsAddr = (unsigned)(uintptr_t)sb;
        uv4 g0;
        g0[0] = 1u;                                    // count=1, is_restore=0, gather off
        g0[1] = ldsAddr;                               // lds_addr [63:32]
        g0[2] = (unsigned)ga;                          // global_addr [95:64]
        g0[3] = (unsigned)(ga >> 32) | (2u << 30);     // global_addr[56:32] | type=2
        uv8 g1;
        g1[0] = 2u << 16;                              // wg_mask=0, data_size=4B
        g1[1] = ((unsigned)BOXF & 0xFFFFu) << 16;      // tensor_dim0[15:0] @ [79:64... (word1 hi16)]
        g1[2] = ((unsigned)BOXF >> 16) | (1u << 16);   // tensor_dim0 hi16 | tensor_dim1 lo16 = 1
        g1[3] = ((unsigned)BOXF) << 16;                // tensor_dim1 hi16=0 | tile_dim0 = 24000
        g1[4] = 1u;                                    // tile_dim1=1, tile_dim2=0
        g1[5] = (unsigned)BOXF;                        // tensor_dim0_stride lo32
        g1[6] = 0u;                                    // dim0_stride hi16 | dim1_stride lo16
        g1[7] = 0u;                                    // dim1_stride hi32
        asm volatile("tensor_load_to_lds %0, %1"
                     :: "s"(g0), "s"(g1)
                     : "memory");
        __builtin_amdgcn_s_wait_tensorcnt(0);
    }
    __syncthreads();

    for (int i = tid; i < KPRE; i += bd) {
        float y1 = sb[i * 4 + 0], x1 = sb[i * 4 + 1];
        float y2 = sb[i * 4 + 2], x2 = sb[i * 4 + 3];
        sa[i] = (y2 - y1) * (x2 - x1);
    }
    for (int i = tid; i < MWORDS; i += bd) mask[i] = 0u;
    __syncthreads();

    int kept = 0;
    for (int i = 0; i < KPRE && kept < PROP; ++i) {
        if ((mask[i >> 5] >> (i & 31)) & 1u) continue;   // uniform read -> uniform branch
        if (tid == 0) keptIdx[kept] = (unsigned short)i;
        kept++;
        float by1 = sb[i * 4 + 0], bx1 = sb[i * 4 + 1];
        float by2 = sb[i * 4 + 2], bx2 = sb[i * 4 + 3];
        float ba  = sa[i];
        for (int j = i + 1 + tid; j < KPRE; j += bd) {
            if ((mask[j >> 5] >> (j & 31)) & 1u) continue;
            float iy1 = fmaxf(by1, sb[j * 4 + 0]);
            float ix1 = fmaxf(bx1, sb[j * 4 + 1]);
            float iy2 = fminf(by2, sb[j * 4 + 2]);
            float ix2 = fminf(bx2, sb[j * 4 + 3]);
            float inter = fmaxf(iy2 - iy1, 0.f) * fmaxf(ix2 - ix1, 0.f);
            float iou = inter / (ba + sa[j] - inter + 1e-12f);
            if (iou > NMS_T) atomicOr(&mask[j >> 5], 1u << (j & 31));
        }
        __syncthreads();
    }
    __syncthreads();

    for (int p = tid; p < PROP; p += bd) {
        float o0 = 0.f, o1 = 0.f, o2 = 0.f, o3 = 0.f;
        if (p < kept) {
            int ii = keptIdx[p];
            o0 = sb[ii * 4 + 0]; o1 = sb[ii * 4 + 1];
            o2 = sb[ii * 4 + 2]; o3 = sb[ii * 4 + 3];
        }
        size_t o = ((size_t)b * PROP + p) * 4;
        out[o + 0] = o0; out[o + 1] = o1; out[o + 2] = o2; out[o + 3] = o3;
    }
}

// ------------------------------------------------------------------- launcher
extern "C" void kernel_launch(void* const* d_in, const int* in_sizes, int n_in,
                              void* d_out, int out_size, void* d_ws, size_t ws_size,
                              hipStream_t stream) {
    const float* probs   = (const float*)d_in[0];   // [B,N,2]
    const float* deltas  = (const float*)d_in[1];   // [B,N,4]
    const float* anchors = (const float*)d_in[2];   // [B,N,4]
    float* out = (float*)d_out;                     // [B,PROP,4]

    int B = out_size / (PROP * 4);
    int N = in_sizes[0] / (2 * B);

    // ws layout: hist1 | hist2 | ctrs | pairs | boxes
    uint8_t* ws = (uint8_t*)d_ws;
    size_t hist1B = (size_t)B * NBKT * 4;
    size_t hist2B = (size_t)B * NBKT * 4;
    size_t ctrB   = (size_t)B * 8 * 4;
    size_t pairsB = (size_t)B * SORTN * 8;
    unsigned* hist1 = (unsigned*)ws;
    unsigned* hist2 = (unsigned*)(ws + hist1B);
    unsigned* ctrs  = (unsigned*)(ws + hist1B + hist2B);
    unsigned long long* pairs = (unsigned long long*)(ws + hist1B + hist2B + ctrB);
    float* boxes = (float*)(ws + hist1B + hist2B + ctrB + pairsB);

    size_t zeroWords = (hist1B + hist2B + ctrB + pairsB) / 4;
    k_zero<<<(unsigned)((zeroWords + 255) / 256), 256, 0, stream>>>((unsigned*)ws, zeroWords);

    dim3 gperN((unsigned)((N + 255) / 256), (unsigned)B);
    k_hist1 <<<gperN, 256, 0, stream>>>(probs, hist1, N);
    k_select1<<<(unsigned)B, 32, 0, stream>>>(hist1, ctrs);
    k_hist2 <<<gperN, 256, 0, stream>>>(probs, ctrs, hist2, N);
    k_select2<<<(unsigned)B, 32, 0, stream>>>(hist2, ctrs);
    k_compact<<<gperN, 256, 0, stream>>>(probs, ctrs, pairs, N);

    // large dynamic LDS (CDNA5 WGP has 320 KB): 64 KB sort, 120000 B NMS
    hipFuncSetAttribute((const void*)k_sort_decode,
                        hipFuncAttributeMaxDynamicSharedMemorySize, SORTN * 8);
    hipFuncSetAttribute((const void*)k_nms,
                        hipFuncAttributeMaxDynamicSharedMemorySize, KPRE * 5 * 4);

    k_sort_decode<<<(unsigned)B, 512, SORTN * 8, stream>>>(pairs, deltas, anchors, boxes, N);
    k_nms<<<(unsigned)B, 256, KPRE * 5 * 4, stream>>>(boxes, out);
}